// OffsetLoss_23536420782515
// MI455X (gfx1250) — compile-verified
//
#include <hip/hip_runtime.h>
#include <hip/hip_bf16.h>

// Problem geometry (fixed by the reference)
#define BDIM   16
#define CDIM   4
#define HDIM   512
#define WDIM   1024
#define NROWS  (BDIM * CDIM * HDIM)   // 32768 rows
#define NBC    (BDIM * CDIM)          // 64 (b,c) accumulators
#define WAVES_PER_BLOCK 8             // 256 threads/block = 8 wave32s

typedef float v4f __attribute__((ext_vector_type(4)));
typedef int   v4i __attribute__((vector_size(4 * sizeof(int))));

// CDNA5 async global->LDS engine (round-1 diagnostic confirmed the builtin and
// its signature: (AS1 v4i*, AS3 v4i*, imm offset, imm cpol)).
#if __has_builtin(__builtin_amdgcn_global_load_async_to_lds_b128) && \
    __has_builtin(__builtin_amdgcn_s_wait_asynccnt)
#define USE_ASYNC_LDS 1
typedef __attribute__((address_space(1))) v4i* gv4p;
typedef __attribute__((address_space(3))) v4i* lv4p;
#else
#define USE_ASYNC_LDS 0
#endif

// Defined FIRST so the disasm snippet shows this kernel's body (verify
// global_load_async_to_lds_b128 + s_wait_asynccnt are emitted).
__global__ __launch_bounds__(WAVES_PER_BLOCK * 32) void row_argmax_kernel(
    const float* __restrict__ offset_pred,
    const float* __restrict__ offset_true,
    const float* __restrict__ cls_true,
    const float* __restrict__ vertical_true,
    float* __restrict__ ws_sum,
    float* __restrict__ ws_cnt) {
  const int wave = threadIdx.x >> 5;
  const int lane = threadIdx.x & 31;
  const int row  = blockIdx.x * WAVES_PER_BLOCK + wave;   // [0, NROWS)
  const size_t base = (size_t)row * WDIM;

  float bv = -__builtin_inff();
  int   bi = 0;

#if USE_ASYNC_LDS
  // Stage this wave's 4KB row into LDS with the async DMA path (ASYNCcnt),
  // then scan from LDS. 8 waves * 4KB = 32KB/block (WGP has 320KB).
  __shared__ __align__(16) float tile[WAVES_PER_BLOCK * WDIM];
  {
    gv4p g = (gv4p)(cls_true + base);          // C-style cast: drops const, AS1
    lv4p l = (lv4p)(tile + wave * WDIM);       // addrspacecast generic -> AS3
#pragma unroll
    for (int k = 0; k < 8; ++k) {
      const int vec = lane + 32 * k;           // coalesced 512B per step
      __builtin_amdgcn_global_load_async_to_lds_b128(g + vec, l + vec, 0, 0);
    }
    __builtin_amdgcn_s_wait_asynccnt(0);
  }
  const v4f* __restrict__ src = (const v4f*)(tile + wave * WDIM);
#pragma unroll
  for (int k = 0; k < 8; ++k) {
    const v4f v = src[lane + 32 * k];
    const int idx0 = (lane + 32 * k) * 4;      // per-lane indices strictly increase
    if (v.x > bv) { bv = v.x; bi = idx0 + 0; }
    if (v.y > bv) { bv = v.y; bi = idx0 + 1; }
    if (v.z > bv) { bv = v.z; bi = idx0 + 2; }
    if (v.w > bv) { bv = v.w; bi = idx0 + 3; }
  }
#else
  // Direct streaming path: non-temporal b128 loads (data is used exactly once).
  const v4f* __restrict__ src = (const v4f*)(cls_true + base);
#pragma unroll
  for (int k = 0; k < 8; ++k) {
    const v4f v = __builtin_nontemporal_load(&src[lane + 32 * k]);
    const int idx0 = (lane + 32 * k) * 4;
    if (v.x > bv) { bv = v.x; bi = idx0 + 0; }
    if (v.y > bv) { bv = v.y; bi = idx0 + 1; }
    if (v.z > bv) { bv = v.z; bi = idx0 + 2; }
    if (v.w > bv) { bv = v.w; bi = idx0 + 3; }
  }
#endif

  // Wave32 (val, idx) reduction; ties resolve to the smallest index -> exact
  // first-max semantics of jnp.argmax / torch.
#pragma unroll
  for (int off = 16; off > 0; off >>= 1) {
    const float ov = __shfl_xor(bv, off, 32);
    const int   oi = __shfl_xor(bi, off, 32);
    if (ov > bv || (ov == bv && oi < bi)) { bv = ov; bi = oi; }
  }

  // Lane 0 gathers the single selected element of each offset tensor.
  __shared__ float serr[WAVES_PER_BLOCK];
  __shared__ float smsk[WAVES_PER_BLOCK];
  if (lane == 0) {
    const float p = offset_pred[base + bi];
    const float t = offset_true[base + bi];
    const float m = (vertical_true[row] >= 0.5f) ? 1.0f : 0.0f;
    serr[wave] = fabsf(p - t) * m;
    smsk[wave] = m;
  }
  __syncthreads();

  // All 8 rows in this block share one (b,c) bucket (8 divides H=512).
  if (threadIdx.x == 0) {
    float e = 0.0f, c = 0.0f;
#pragma unroll
    for (int i = 0; i < WAVES_PER_BLOCK; ++i) { e += serr[i]; c += smsk[i]; }
    const int bc = row >> 9;  // row / HDIM
    atomicAdd(&ws_sum[bc], e);
    atomicAdd(&ws_cnt[bc], c);
  }
}

__global__ __launch_bounds__(2 * NBC) void zero_ws_kernel(float* ws) {
  ws[threadIdx.x] = 0.0f;  // ws[0..63]=err sums, ws[64..127]=mask counts
}

__global__ __launch_bounds__(64) void finalize_kernel(const float* __restrict__ ws_sum,
                                                      const float* __restrict__ ws_cnt,
                                                      float* __restrict__ out) {
  const int i = threadIdx.x;  // 64 threads = 64 (b,c) buckets
  const float c = ws_cnt[i];
  float per = (c > 0.0f) ? (ws_sum[i] / c) : 0.0f;
#pragma unroll
  for (int off = 16; off > 0; off >>= 1) per += __shfl_xor(per, off, 32);
  __shared__ float acc[2];
  if ((i & 31) == 0) acc[i >> 5] = per;
  __syncthreads();
  if (i == 0) out[0] = (acc[0] + acc[1]) * (1.0f / (float)BDIM);
}

extern "C" void kernel_launch(void* const* d_in, const int* in_sizes, int n_in,
                              void* d_out, int out_size, void* d_ws, size_t ws_size,
                              hipStream_t stream) {
  const float* offset_pred   = (const float*)d_in[0];
  const float* offset_true   = (const float*)d_in[1];
  const float* cls_true      = (const float*)d_in[2];
  const float* vertical_true = (const float*)d_in[3];

  float* ws_sum = (float*)d_ws;        // 64 floats
  float* ws_cnt = ws_sum + NBC;        // 64 floats

  zero_ws_kernel<<<1, 2 * NBC, 0, stream>>>(ws_sum);
  row_argmax_kernel<<<NROWS / WAVES_PER_BLOCK, WAVES_PER_BLOCK * 32, 0, stream>>>(
      offset_pred, offset_true, cls_true, vertical_true, ws_sum, ws_cnt);
  finalize_kernel<<<1, 64, 0, stream>>>(ws_sum, ws_cnt, (float*)d_out);
}